// Auto_attn_66331474920165
// MI455X (gfx1250) — compile-verified
//
#include <hip/hip_runtime.h>
#include <cstdint>
#include <math.h>

// Shapes: values/corr (B=32, H=8, C=64, L=1024) fp32; top_k = 6 (pad to 8).
#define B_N 32
#define HC_N 512
#define L_N 1024
#define KTOP 6

typedef __attribute__((ext_vector_type(2))) float v2f;
typedef __attribute__((ext_vector_type(8))) float v8f;

// ---------------------------------------------------------------------------
// Phase 1: mean over (H,C):  mv[b,l] = (1/512) * sum_hc corr[b,hc,l]
// grid (32, 4, 4): b, l-chunk(256), hc-chunk(128). Coalesced over l.
// ---------------------------------------------------------------------------
__global__ __launch_bounds__(256)
void ac_mean_hc(const float* __restrict__ corr, float* __restrict__ mv) {
    const int b   = blockIdx.x;
    const int l   = blockIdx.y * 256 + threadIdx.x;
    const int hc0 = blockIdx.z * 128;
    const float* base = corr + ((size_t)b * HC_N) * L_N + l;
    float s = 0.f;
    #pragma unroll 4
    for (int hc = hc0; hc < hc0 + 128; ++hc)
        s += base[(size_t)hc * L_N];
    atomicAdd(&mv[b * L_N + l], s * (1.0f / 512.0f));
}

// ---------------------------------------------------------------------------
// Phase 2: global mean over b, top-6 (tie -> lowest index, matching lax.top_k),
// then per-batch softmax of gathered weights. One block of 1024 threads.
// ---------------------------------------------------------------------------
__global__ __launch_bounds__(1024)
void ac_topk_softmax(const float* __restrict__ mv, int* __restrict__ idx8,
                     float* __restrict__ w8) {
    __shared__ float rv[L_N];
    __shared__ int   ri[L_N];
    __shared__ int   winner;
    __shared__ int   topIdx[8];
    const int tid = threadIdx.x;

    float g = 0.f;
    #pragma unroll 8
    for (int b = 0; b < B_N; ++b) g += mv[b * L_N + tid];

    for (int k = 0; k < KTOP; ++k) {
        rv[tid] = g; ri[tid] = tid;
        __syncthreads();
        for (int s = 512; s > 0; s >>= 1) {
            if (tid < s) {
                float a = rv[tid], c = rv[tid + s];
                int ia = ri[tid], ic = ri[tid + s];
                if (c > a || (c == a && ic < ia)) { rv[tid] = c; ri[tid] = ic; }
            }
            __syncthreads();
        }
        if (tid == 0) { winner = ri[0]; topIdx[k] = ri[0]; }
        __syncthreads();
        if (tid == winner) g = -INFINITY;
        __syncthreads();
    }
    if (tid == 0) { topIdx[6] = 0; topIdx[7] = 0; }
    __syncthreads();
    if (tid < 8) idx8[tid] = topIdx[tid];
    if (tid < B_N) {
        const int b = tid;
        float w[KTOP], m = -INFINITY;
        #pragma unroll
        for (int k = 0; k < KTOP; ++k) { w[k] = mv[b * L_N + topIdx[k]]; m = fmaxf(m, w[k]); }
        float den = 0.f;
        #pragma unroll
        for (int k = 0; k < KTOP; ++k) { w[k] = expf(w[k] - m); den += w[k]; }
        const float inv = 1.0f / den;
        #pragma unroll
        for (int k = 0; k < KTOP; ++k) w8[b * 8 + k] = w[k] * inv;
        w8[b * 8 + 6] = 0.f; w8[b * 8 + 7] = 0.f;   // pad weights -> zero contribution
    }
}

// ---------------------------------------------------------------------------
// Phase 3: out[b,h,c,l] = sum_k w[b,k] * row[(l+idx_k)&1023]
// One block per (b,h,c) row. Row (4KB) staged in LDS with
// global_load_async_to_lds_b128 (256 lanes x 16B). Reduction over k done with
// chained V_WMMA_F32_16X16X4_F32: A = weights (replicated over M),
// B = gathered row values (N = 16 consecutive l). D rows are identical, so
// d[0] of lanes 0..15 (tile at l0) and lanes 16..31 (tile at l0+16) give a
// fully coalesced 32-wide store of out[l0..l0+31].
// ---------------------------------------------------------------------------
__global__ __launch_bounds__(256)
void ac_delay_agg(const float* __restrict__ values, const int* __restrict__ idx8,
                  const float* __restrict__ w8, float* __restrict__ out) {
    __shared__ float row[L_N];
    const int tid = threadIdx.x;
    const int bhc = blockIdx.x;
    const int b   = bhc >> 9;                 // / (H*C)
    const size_t rowbase = (size_t)bhc * L_N;

    // Async DMA global -> LDS (gfx1250 path, tracked by ASYNCcnt)
    uint32_t ldsoff = (uint32_t)(uintptr_t)(&row[0]) + (uint32_t)tid * 16u;
    uint64_t ga     = (uint64_t)(uintptr_t)(values + rowbase) + (uint64_t)tid * 16u;
    asm volatile("global_load_async_to_lds_b128 %0, %1, off"
                 :: "v"(ldsoff), "v"(ga) : "memory");

    // Overlap: fetch weights/indices while the async copy is in flight
    int   idx[8];
    float w[8];
    #pragma unroll
    for (int k = 0; k < 8; ++k) { idx[k] = idx8[k]; w[k] = w8[b * 8 + k]; }

    asm volatile("s_wait_asynccnt 0x0" ::: "memory");
    __syncthreads();

    const int  lane = tid & 31;
    const int  wv   = tid >> 5;
    const int  m    = lane & 15;
    const bool hi   = lane >= 16;

    // A-matrix (16x4 f32): v0 = {K0|K2}, v1 = {K1|K3}; weights replicated over M
    v2f a0, a1;
    a0.x = hi ? w[2] : w[0];  a0.y = hi ? w[3] : w[1];
    a1.x = hi ? w[6] : w[4];  a1.y = hi ? w[7] : w[5];
    const int i0x = hi ? idx[2] : idx[0], i0y = hi ? idx[3] : idx[1];
    const int i1x = hi ? idx[6] : idx[4], i1y = hi ? idx[7] : idx[5];

    #pragma unroll
    for (int it = 0; it < 4; ++it) {
        const int l0  = wv * 128 + it * 32;
        const int lt0 = l0 + m;          // tile 0: columns l0 .. l0+15
        const int lt1 = l0 + 16 + m;     // tile 1: columns l0+16 .. l0+31

        // B-matrix (4x16 f32): v0 = rows {K0|K2}, v1 = rows {K1|K3}
        v2f b00, b01, b10, b11;
        b00.x = row[(lt0 + i0x) & (L_N - 1)];  b00.y = row[(lt0 + i0y) & (L_N - 1)];
        b01.x = row[(lt0 + i1x) & (L_N - 1)];  b01.y = row[(lt0 + i1y) & (L_N - 1)];
        b10.x = row[(lt1 + i0x) & (L_N - 1)];  b10.y = row[(lt1 + i0y) & (L_N - 1)];
        b11.x = row[(lt1 + i1x) & (L_N - 1)];  b11.y = row[(lt1 + i1y) & (L_N - 1)];

        v8f acc0 = {}; v8f acc1 = {};
        acc0 = __builtin_amdgcn_wmma_f32_16x16x4_f32(false, a0, false, b00, (short)0, acc0, false, false);
        acc0 = __builtin_amdgcn_wmma_f32_16x16x4_f32(false, a1, false, b01, (short)0, acc0, false, false);
        acc1 = __builtin_amdgcn_wmma_f32_16x16x4_f32(false, a0, false, b10, (short)0, acc1, false, false);
        acc1 = __builtin_amdgcn_wmma_f32_16x16x4_f32(false, a1, false, b11, (short)0, acc1, false, false);

        // lanes 0-15: tile0 (N=lane) ; lanes 16-31: tile1 (N=lane-16) -> l = l0+lane
        const float val = hi ? acc1[0] : acc0[0];
        out[rowbase + l0 + lane] = val;
    }
}

// ---------------------------------------------------------------------------
extern "C" void kernel_launch(void* const* d_in, const int* in_sizes, int n_in,
                              void* d_out, int out_size, void* d_ws, size_t ws_size,
                              hipStream_t stream) {
    const float* values = (const float*)d_in[0];
    const float* corr   = (const float*)d_in[1];
    float*       out    = (float*)d_out;

    char*  ws   = (char*)d_ws;
    float* mv   = (float*)ws;                                   // 32*1024 f32
    int*   idx8 = (int*)(ws + (size_t)B_N * L_N * sizeof(float)); // 8 ints
    float* w8   = (float*)(ws + (size_t)B_N * L_N * sizeof(float) + 32); // 32*8 f32

    hipMemsetAsync(mv, 0, (size_t)B_N * L_N * sizeof(float), stream);

    ac_mean_hc<<<dim3(B_N, 4, 4), 256, 0, stream>>>(corr, mv);
    ac_topk_softmax<<<1, 1024, 0, stream>>>(mv, idx8, w8);
    ac_delay_agg<<<B_N * HC_N, 256, 0, stream>>>(values, idx8, w8, out);
}